// ConditionalChannelWeighting_59468117180562
// MI455X (gfx1250) — compile-verified
//
#include <hip/hip_runtime.h>
#include <math.h>

// ============================================================================
// ConditionalChannelWeighting (Lite-HRNet block) for MI455X (gfx1250).
//
// Memory-bound: ~157MB min traffic -> ~7us floor at 23.3 TB/s. All heavy
// tensors are touched once (fused CRW-multiply into dwconv; fused SE-scale
// into the channel-shuffle writeback). The CRW 1x1-conv GEMMs
// ([4096x300]x[300x37] and [4096x40]x[40x300]) run on V_WMMA_F32_16X16X4_F32
// in full fp32 precision, one wave per 16-row tile, LDS re-layout between
// the two GEMMs. Padding guards are expressed as clamped-index loads times
// precomputed 0/1 lane masks so the GEMM inner loops are branch-free and
// EXEC stays all-ones (WMMA requirement) with clause-able vector loads.
// ============================================================================

typedef float v2f __attribute__((ext_vector_type(2)));
typedef float v8f __attribute__((ext_vector_type(8)));

#define BN_EPS 1e-5f
#define NB 64          // batch
#define SUMC 300       // concat channels (20+40+80+160)
#define MIDC 37        // SUM/8
#define MROWS 4096     // NB * 8 * 8

static __device__ __forceinline__ float sigmoidf_(float x) {
    return 1.0f / (1.0f + __expf(-x));
}

static __device__ __forceinline__ int imin_(int a, int b) { return a < b ? a : b; }

// ---------------------------------------------------------------------------
// K1: build pooled concat P[B][300][8][8] from the second-half channels of
// each branch (adaptive avg-pool to 8x8; branch 3 is identity).
// ---------------------------------------------------------------------------
__global__ void pool_concat_kernel(const float* __restrict__ x0,
                                   const float* __restrict__ x1,
                                   const float* __restrict__ x2,
                                   const float* __restrict__ x3,
                                   float* __restrict__ P) {
    int idx = blockIdx.x * blockDim.x + threadIdx.x;
    const int total = NB * SUMC * 64;
    if (idx >= total) return;
    int s = idx & 63;
    int t = idx >> 6;
    int cc = t % SUMC;
    int b  = t / SUMC;

    const float* xp; int Cfull, H, c;
    if (cc < 20)       { xp = x0; Cfull = 40;  H = 64; c = cc;       }
    else if (cc < 60)  { xp = x1; Cfull = 80;  H = 32; c = cc - 20;  }
    else if (cc < 140) { xp = x2; Cfull = 160; H = 16; c = cc - 60;  }
    else               { xp = x3; Cfull = 320; H = 8;  c = cc - 140; }
    int f = H >> 3;
    int h8 = s >> 3, w8 = s & 7;
    const float* base = xp + ((size_t)b * Cfull + (Cfull / 2 + c)) * H * H;
    float sum = 0.0f;
    for (int dy = 0; dy < f; ++dy)
        for (int dx = 0; dx < f; ++dx)
            sum += base[(h8 * f + dy) * H + (w8 * f + dx)];
    P[idx] = sum / (float)(f * f);
}

// ---------------------------------------------------------------------------
// K2: CrossResolutionWeighting GEMMs on WMMA fp32 16x16x4.
//   t  = relu(bn1(P @ w1^T + b1))        M=4096, K=300, N=37 (pad 48)
//   At = sigmoid(bn2(t @ w2^T + b2))     M=4096, K=40 (pad), N=300 (pad 304)
// One wave (32 threads) per 16-row tile; 256 blocks.
// A layout (f32 16x4): lanes 0-15 M=0..15 hold K=2*hi in .x, K=2*hi+1 in .y
//   (hi = lane>=16). B layout mirrors with N across lanes.
// C/D layout: element j -> M = j + 8*hi, N = lane&15.
// ---------------------------------------------------------------------------
__global__ __launch_bounds__(32)
void crw_gemm_kernel(const float* __restrict__ P,
                     const float* __restrict__ w1, const float* __restrict__ b1,
                     const float* __restrict__ g1, const float* __restrict__ be1,
                     const float* __restrict__ m1, const float* __restrict__ v1,
                     const float* __restrict__ w2, const float* __restrict__ b2,
                     const float* __restrict__ g2, const float* __restrict__ be2,
                     const float* __restrict__ m2, const float* __restrict__ v2,
                     float* __restrict__ Attn) {
    __shared__ float tbuf[16 * 40];   // t tile, K padded 37 -> 40 with zeros

    const int lane = threadIdx.x;
    const int m    = lane & 15;
    const int hi   = lane >> 4;       // 0 or 1
    const int rowTile = blockIdx.x;   // 0..255
    const int r  = rowTile * 16 + m;  // row this lane loads for A
    const int bb = r >> 6;
    const int ss = r & 63;
    const float* prow = P + (size_t)bb * (SUMC * 64) + ss;   // + c*64

    // zero LDS tile (covers zero-padding of K=37..39)
    for (int i = lane; i < 16 * 40; i += 32) tbuf[i] = 0.0f;
    __syncthreads();

    // Per-lane masks/clamps for the padded third N tile (o = 32+m, valid o<37).
    const float mask2 = (32 + m < MIDC) ? 1.0f : 0.0f;
    const int   o2c   = imin_(32 + m, MIDC - 1);
    const int   o0    = m;
    const int   o1    = 16 + m;

    // ---- GEMM1: K=300 in 75 steps of 4 (branch-free inner loop) ----
    v8f acc0 = {}, acc1 = {}, acc2 = {};
    for (int kb = 0; kb < SUMC; kb += 4) {
        const int k0 = kb + 2 * hi;
        v2f a;
        a.x = prow[(size_t)k0 * 64];
        a.y = prow[(size_t)(k0 + 1) * 64];
        // B tiles: B[k][n] = w1[n][k]; padded tile via clamped load * mask
        v2f bv0, bv1, bv2;
        bv0.x = w1[o0 * SUMC + k0];
        bv0.y = w1[o0 * SUMC + k0 + 1];
        bv1.x = w1[o1 * SUMC + k0];
        bv1.y = w1[o1 * SUMC + k0 + 1];
        bv2.x = w1[o2c * SUMC + k0]     * mask2;
        bv2.y = w1[o2c * SUMC + k0 + 1] * mask2;
        acc0 = __builtin_amdgcn_wmma_f32_16x16x4_f32(false, a, false, bv0,
                                                     (short)0, acc0, false, false);
        acc1 = __builtin_amdgcn_wmma_f32_16x16x4_f32(false, a, false, bv1,
                                                     (short)0, acc1, false, false);
        acc2 = __builtin_amdgcn_wmma_f32_16x16x4_f32(false, a, false, bv2,
                                                     (short)0, acc2, false, false);
    }

    // ---- epilogue 1: bias + BN + ReLU, stage into LDS in A-friendly layout
    // bn(x+b) = x*sc + ((b-m)*sc + be)
    float sc0, sh0, sc1, sh1, sc2, sh2;
    {
        sc0 = g1[o0] * rsqrtf(v1[o0] + BN_EPS);
        sh0 = (b1[o0] - m1[o0]) * sc0 + be1[o0];
    }
    {
        sc1 = g1[o1] * rsqrtf(v1[o1] + BN_EPS);
        sh1 = (b1[o1] - m1[o1]) * sc1 + be1[o1];
    }
    {
        sc2 = g1[o2c] * rsqrtf(v1[o2c] + BN_EPS);
        sh2 = (b1[o2c] - m1[o2c]) * sc2 + be1[o2c];
    }
#pragma unroll
    for (int j = 0; j < 8; ++j) {
        int row = j + 8 * hi;
        tbuf[row * 40 + m]      = fmaxf(acc0[j] * sc0 + sh0, 0.0f);
        tbuf[row * 40 + 16 + m] = fmaxf(acc1[j] * sc1 + sh1, 0.0f);
        if (32 + m < MIDC)
            tbuf[row * 40 + 32 + m] = fmaxf(acc2[j] * sc2 + sh2, 0.0f);
    }
    __syncthreads();

    // ---- GEMM2: K=40 (A zero-padded in LDS), N=300 over 19 tiles of 16 ----
    for (int ptile = 0; ptile < 19; ++ptile) {
        const int   p     = ptile * 16 + m;
        const float maskP = (p < SUMC) ? 1.0f : 0.0f;
        const int   pc    = imin_(p, SUMC - 1);
        v8f acc = {};
        for (int kb = 0; kb < 40; kb += 4) {
            const int k0 = kb + 2 * hi;
            v2f a;
            a.x = tbuf[m * 40 + k0];
            a.y = tbuf[m * 40 + k0 + 1];
            // B[k][n] = w2[p][k], zero for k>=37 or p>=300 (clamped load * mask)
            const float mkx = (k0     < MIDC) ? maskP : 0.0f;
            const float mky = (k0 + 1 < MIDC) ? maskP : 0.0f;
            const int   kx  = imin_(k0,     MIDC - 1);
            const int   ky  = imin_(k0 + 1, MIDC - 1);
            v2f bv;
            bv.x = w2[pc * MIDC + kx] * mkx;
            bv.y = w2[pc * MIDC + ky] * mky;
            acc = __builtin_amdgcn_wmma_f32_16x16x4_f32(false, a, false, bv,
                                                        (short)0, acc, false, false);
        }
        // epilogue 2: bias + BN + sigmoid -> Attn[b][p][s]
        float sc = g2[pc] * rsqrtf(v2[pc] + BN_EPS);
        float sh = (b2[pc] - m2[pc]) * sc + be2[pc];
#pragma unroll
        for (int j = 0; j < 8; ++j) {
            if (p < SUMC) {
                int row = j + 8 * hi;
                int r2  = rowTile * 16 + row;
                int b_  = r2 >> 6, s_ = r2 & 63;
                float v = acc[j] * sc + sh;
                Attn[(size_t)b_ * (SUMC * 64) + p * 64 + s_] = sigmoidf_(v);
            }
        }
    }
}

// ---------------------------------------------------------------------------
// K3: fused CRW-multiply + depthwise 3x3 conv for one branch.
// out[b][c][y][x] = bias[c] + sum_{dy,dx} w[c][dy][dx] *
//                   x2h[b][c][yy][xx] * Attn[b][choff+c][yy/f][xx/f]
// ---------------------------------------------------------------------------
__global__ void dwconv_crw_kernel(const float* __restrict__ x,    // full branch [B][2C][H][H]
                                  const float* __restrict__ attn, // [B][300][8][8]
                                  const float* __restrict__ w,    // [C][3][3]
                                  const float* __restrict__ bias, // [C]
                                  float* __restrict__ out,        // [B][C][H][H]
                                  int C, int H, int choff) {
    int idx = blockIdx.x * blockDim.x + threadIdx.x;
    int total = NB * C * H * H;
    if (idx >= total) return;
    int xw = idx % H;
    int t  = idx / H;
    int y  = t % H; t /= H;
    int c  = t % C;
    int b  = t / C;
    int f  = H >> 3;

    const float* xbase = x + ((size_t)b * (2 * C) + (C + c)) * H * H;
    const float* abase = attn + ((size_t)b * SUMC + (choff + c)) * 64;
    const float* wc    = w + c * 9;

    float acc = bias[c];
    for (int dy = -1; dy <= 1; ++dy) {
        int yy = y + dy;
        if (yy < 0 || yy >= H) continue;
        for (int dx = -1; dx <= 1; ++dx) {
            int xx = xw + dx;
            if (xx < 0 || xx >= H) continue;
            float a = abase[(yy / f) * 8 + (xx / f)];
            acc += wc[(dy + 1) * 3 + (dx + 1)] * xbase[yy * H + xx] * a;
        }
    }
    out[idx] = acc;
}

// ---------------------------------------------------------------------------
// K4: SE global average pool per (b,c): one block per (b,c), LDS reduce.
// ---------------------------------------------------------------------------
__global__ void se_pool_kernel(const float* __restrict__ s, float* __restrict__ pool,
                               int HW) {
    int bc = blockIdx.x;
    const float* p = s + (size_t)bc * HW;
    float sum = 0.0f;
    for (int i = threadIdx.x; i < HW; i += blockDim.x) sum += p[i];
    __shared__ float red[256];
    red[threadIdx.x] = sum;
    __syncthreads();
    for (int st = 128; st > 0; st >>= 1) {
        if ((int)threadIdx.x < st) red[threadIdx.x] += red[threadIdx.x + st];
        __syncthreads();
    }
    if (threadIdx.x == 0) pool[bc] = red[0] / (float)HW;
}

// ---------------------------------------------------------------------------
// K5: SE MLP (ratio 4): one block per batch element.
// ---------------------------------------------------------------------------
__global__ void se_mlp_kernel(const float* __restrict__ pool,
                              const float* __restrict__ w1, const float* __restrict__ b1,
                              const float* __restrict__ w2, const float* __restrict__ b2,
                              float* __restrict__ seattn, int C) {
    int b = blockIdx.x;
    int t = threadIdx.x;
    int mid = C / 4;
    __shared__ float h[64];   // mid <= 40
    const float* pb = pool + (size_t)b * C;
    if (t < mid) {
        float s = b1[t];
        for (int c = 0; c < C; ++c) s += w1[t * C + c] * pb[c];
        h[t] = fmaxf(s, 0.0f);
    }
    __syncthreads();
    if (t < C) {
        float s = b2[t];
        for (int j = 0; j < mid; ++j) s += w2[t * mid + j] * h[j];
        seattn[(size_t)b * C + t] = sigmoidf_(s);
    }
}

// ---------------------------------------------------------------------------
// K6: fused SE-scale + concat + channel_shuffle(groups=2) writeback.
// out[b][2j]   = x[b][j]            (first-half passthrough)
// out[b][2j+1] = s[b][j] * se[b][j]
// ---------------------------------------------------------------------------
__global__ void shuffle_out_kernel(const float* __restrict__ x,   // [B][2C][H][H]
                                   const float* __restrict__ s,   // [B][C][H][H]
                                   const float* __restrict__ se,  // [B][C]
                                   float* __restrict__ out,       // [B][2C][H][H]
                                   int C, int H) {
    int idx = blockIdx.x * blockDim.x + threadIdx.x;
    int HW = H * H;
    int total = NB * 2 * C * HW;
    if (idx >= total) return;
    int sp = idx % HW;
    int t  = idx / HW;
    int co = t % (2 * C);
    int b  = t / (2 * C);
    int j  = co >> 1;
    float v;
    if ((co & 1) == 0)
        v = x[((size_t)b * 2 * C + j) * HW + sp];
    else
        v = s[((size_t)b * C + j) * HW + sp] * se[(size_t)b * C + j];
    out[idx] = v;
}

// ---------------------------------------------------------------------------
static inline int cdiv(int a, int b) { return (a + b - 1) / b; }

extern "C" void kernel_launch(void* const* d_in, const int* in_sizes, int n_in,
                              void* d_out, int out_size, void* d_ws, size_t ws_size,
                              hipStream_t stream) {
    const int BR[4]    = {20, 40, 80, 160};
    const int SIZES[4] = {64, 32, 16, 8};
    const int CHOFF[4] = {0, 20, 60, 140};

    // inputs (dict order, lists tree-flattened)
    const float* x[4];
    for (int i = 0; i < 4; ++i) x[i] = (const float*)d_in[i];
    const float* crw_w1  = (const float*)d_in[4];
    const float* crw_b1  = (const float*)d_in[5];
    const float* crw_g1  = (const float*)d_in[6];
    const float* crw_be1 = (const float*)d_in[7];
    const float* crw_m1  = (const float*)d_in[8];
    const float* crw_v1  = (const float*)d_in[9];
    const float* crw_w2  = (const float*)d_in[10];
    const float* crw_b2  = (const float*)d_in[11];
    const float* crw_g2  = (const float*)d_in[12];
    const float* crw_be2 = (const float*)d_in[13];
    const float* crw_m2  = (const float*)d_in[14];
    const float* crw_v2  = (const float*)d_in[15];
    const float *dw_w[4], *dw_b[4], *sw_w1[4], *sw_b1[4], *sw_w2[4], *sw_b2[4];
    for (int i = 0; i < 4; ++i) {
        dw_w[i]  = (const float*)d_in[16 + i];
        dw_b[i]  = (const float*)d_in[20 + i];
        sw_w1[i] = (const float*)d_in[24 + i];
        sw_b1[i] = (const float*)d_in[28 + i];
        sw_w2[i] = (const float*)d_in[32 + i];
        sw_b2[i] = (const float*)d_in[36 + i];
    }

    // workspace carve-up (floats)
    float* ws = (float*)d_ws;
    float* P    = ws;               ws += NB * SUMC * 64;          // 1,228,800
    float* Attn = ws;               ws += NB * SUMC * 64;          // 1,228,800
    float* S[4];
    for (int i = 0; i < 4; ++i) { S[i] = ws; ws += NB * BR[i] * SIZES[i] * SIZES[i]; }
    float* POOL[4];
    for (int i = 0; i < 4; ++i) { POOL[i] = ws; ws += NB * BR[i]; }
    float* SEAT[4];
    for (int i = 0; i < 4; ++i) { SEAT[i] = ws; ws += NB * BR[i]; }

    // output offsets (branch tensors concatenated flat)
    float* out = (float*)d_out;
    size_t ooff[4];
    size_t acc = 0;
    for (int i = 0; i < 4; ++i) {
        ooff[i] = acc;
        acc += (size_t)NB * 2 * BR[i] * SIZES[i] * SIZES[i];
    }

    // K1: pooled concat
    {
        int n = NB * SUMC * 64;
        pool_concat_kernel<<<cdiv(n, 256), 256, 0, stream>>>(x[0], x[1], x[2], x[3], P);
    }
    // K2: CRW attention via WMMA fp32 GEMMs
    crw_gemm_kernel<<<MROWS / 16, 32, 0, stream>>>(
        P, crw_w1, crw_b1, crw_g1, crw_be1, crw_m1, crw_v1,
        crw_w2, crw_b2, crw_g2, crw_be2, crw_m2, crw_v2, Attn);
    // K3: fused CRW-multiply + depthwise conv per branch
    for (int i = 0; i < 4; ++i) {
        int n = NB * BR[i] * SIZES[i] * SIZES[i];
        dwconv_crw_kernel<<<cdiv(n, 256), 256, 0, stream>>>(
            x[i], Attn, dw_w[i], dw_b[i], S[i], BR[i], SIZES[i], CHOFF[i]);
    }
    // K4/K5: SE per branch
    for (int i = 0; i < 4; ++i) {
        int HW = SIZES[i] * SIZES[i];
        se_pool_kernel<<<NB * BR[i], 256, 0, stream>>>(S[i], POOL[i], HW);
        se_mlp_kernel<<<NB, BR[i], 0, stream>>>(POOL[i], sw_w1[i], sw_b1[i],
                                                sw_w2[i], sw_b2[i], SEAT[i], BR[i]);
    }
    // K6: fused SE-scale + concat + channel shuffle
    for (int i = 0; i < 4; ++i) {
        int n = NB * 2 * BR[i] * SIZES[i] * SIZES[i];
        shuffle_out_kernel<<<cdiv(n, 256), 256, 0, stream>>>(
            x[i], S[i], SEAT[i], out + ooff[i], BR[i], SIZES[i]);
    }
}